// SpatialRPE_17197049053733
// MI455X (gfx1250) — compile-verified
//
#include <hip/hip_runtime.h>

// CDNA5 / gfx1250, wave32. Matrix ops via V_WMMA_F32_16X16X4_F32 (exact fp32).
typedef __attribute__((ext_vector_type(2))) float v2f;
typedef __attribute__((ext_vector_type(8))) float v8f;

#define N_PTS 1024
#define LDS_STRIDE 68   // 64 + pad: avoids LDS bank conflicts on row accesses

// ds_swizzle_b32 group-of-32 XOR patterns: offset = (xor<<10) | and_mask(0x1f)
#define SWZ_X1 0x041F
#define SWZ_X2 0x081F
#define SWZ_X4 0x101F
#define SWZ_X8 0x201F
// lane-xor add-reduce step with immediate pattern (no address VALU, no cndmask)
#define SWZ_ADD(x, pat) \
    (x) += __int_as_float(__builtin_amdgcn_ds_swizzle(__float_as_int(x), (pat)))

// ---------------------------------------------------------------------------
// Pass 1: per-point precompute (B*N = 2048 points):
//   table[n] = (start.x, start.y, cos(heading), sin(heading))
// ---------------------------------------------------------------------------
__global__ void rpe_precompute(const float* __restrict__ road,
                               float4* __restrict__ table, int total) {
    int n = blockIdx.x * blockDim.x + threadIdx.x;
    if (n >= total) return;
    float4 rv = ((const float4*)road)[n];
    float h = atan2f(rv.w - rv.y, rv.z - rv.x);   // heading of (end-start)
    float4 t;
    t.x = rv.x; t.y = rv.y; t.z = cosf(h); t.w = sinf(h);
    table[n] = t;
}

// ---------------------------------------------------------------------------
// Pass 2: one wave32 per 16-row j-tile of a fixed (b, i).
//   GEMM1 [16x5]x[5x64] and GEMM2 [16x64]x[64x8->16] on the matrix pipe.
// PRE=true : read precomputed table from d_ws (no per-pair trig).
// PRE=false: fallback, compute trig inline from road vectors.
// ---------------------------------------------------------------------------
template <bool PRE>
__global__ __launch_bounds__(256) void spatial_rpe_kernel(
    const float* __restrict__ src,     // PRE ? table[B*N*4] : road[B,N,4]
    const float* __restrict__ W1,      // [5,64]
    const float* __restrict__ b1,      // [64]
    const float* __restrict__ lng,     // [64]
    const float* __restrict__ lnb,     // [64]
    const float* __restrict__ W2,      // [64,8]
    const float* __restrict__ b2,      // [8]
    float* __restrict__ out)           // [B,8,1024,1024]
{
    __shared__ float lds[8][16 * LDS_STRIDE];

    const int lane = threadIdx.x & 31;
    const int wave = threadIdx.x >> 5;
    const int m    = lane & 15;   // WMMA row (A) / column (B,C,D)
    const int half = lane >> 4;   // selects K-pair (A/B) or row+8 (C/D)
    const int mm   = m & 7;       // clamped head index (always in-bounds)

    // ---- preload all weights unconditionally (in-bounds for every lane) ----
    float w1a[4], w1b[4], w1c[4], b1v[4], gv[4], bev[4];
#pragma unroll
    for (int t = 0; t < 4; ++t) {
        const int col = t * 16 + m;
        w1a[t] = W1[(half ? 2 : 0) * 64 + col];
        w1b[t] = W1[(half ? 3 : 1) * 64 + col];
        w1c[t] = half ? 0.0f : W1[4 * 64 + col];   // K=4 row, upper half zero
        b1v[t] = b1[col];
        gv[t]  = lng[col];
        bev[t] = lnb[col];
    }
    const float hmask = (m < 8) ? 1.0f : 0.0f;     // zero pad-heads by value
    float w2a[16], w2b[16];
#pragma unroll
    for (int k = 0; k < 16; ++k) {
        const int k0 = 4 * k + 2 * half;
        w2a[k] = W2[k0 * 8 + mm] * hmask;
        w2b[k] = W2[(k0 + 1) * 8 + mm] * hmask;
    }
    const float b2v = b2[mm];

    // tile id: [b(1) | i(10) | jt(6)] ; total 2*1024*64 = 131072 tiles
    const unsigned tile = blockIdx.x * 8u + wave;
    const int b   = tile >> 16;
    const int rem = tile & 0xFFFF;
    const int i   = rem >> 6;
    const int jt  = rem & 63;
    const int j   = jt * 16 + m;

    // ---- per-pair geometric features ----
    float dx, dy, ci, si, f1, f2;
    if (PRE) {
        const float4 ti = ((const float4*)src)[b * N_PTS + i];  // (sx,sy,c,s)
        const float4 tj = ((const float4*)src)[b * N_PTS + j];
        dx = ti.x - tj.x;  dy = ti.y - tj.y;
        ci = ti.z;  si = ti.w;
        f1 = ci * tj.z + si * tj.w;    // cos(hi-hj)
        f2 = si * tj.z - ci * tj.w;    // sin(hi-hj)
    } else {
        const float4 rvj = ((const float4*)src)[b * N_PTS + j];
        const float4 rvi = ((const float4*)src)[b * N_PTS + i];
        const float hj = atan2f(rvj.w - rvj.y, rvj.z - rvj.x);
        const float hi = atan2f(rvi.w - rvi.y, rvi.z - rvi.x);
        dx = rvi.x - rvj.x;  dy = rvi.y - rvj.y;
        ci = cosf(hi);  si = sinf(hi);
        f1 = cosf(hi - hj);  f2 = sinf(hi - hj);
    }
    const float f0 = __logf(sqrtf(dx * dx + dy * dy) + 1e-5f);  // feat_dist
    const float f3 =  dx * ci + dy * si;                        // feat_rel_x
    const float f4 = -dx * si + dy * ci;                        // feat_rel_y

    // A 16x4 f32 layout: lanes0-15 {v0=K0,v1=K1}, lanes16-31 {v0=K2,v1=K3}
    v2f a1, a2;
    a1.x = half ? f2 : f0;
    a1.y = half ? f3 : f1;
    a2.x = half ? 0.f : f4;   // K=4 feature, zero-padded to K=8
    a2.y = 0.f;

    // ---- GEMM1: h[16x64] = geom[16x5] * W1[5x64] + b1 ----
    v8f c[4];
#pragma unroll
    for (int t = 0; t < 4; ++t) {
        v2f bm1; bm1.x = w1a[t]; bm1.y = w1b[t];
        v2f bm2; bm2.x = w1c[t]; bm2.y = 0.f;
        v8f acc = {};
        acc = __builtin_amdgcn_wmma_f32_16x16x4_f32(false, a1, false, bm1,
                                                    (short)0, acc, false, false);
        acc = __builtin_amdgcn_wmma_f32_16x16x4_f32(false, a2, false, bm2,
                                                    (short)0, acc, false, false);
#pragma unroll
        for (int r = 0; r < 8; ++r) acc[r] += b1v[t];
        c[t] = acc;
    }

    // ---- LayerNorm over 64 cols per row (row = vgpr r + 8*half) ----
    // lane-XOR tree via ds_swizzle immediates; masks 1..8 stay in 16-lane half
    float mean[8], rstd[8];
#pragma unroll
    for (int r = 0; r < 8; ++r) {
        float s = c[0][r] + c[1][r] + c[2][r] + c[3][r];
        float q = c[0][r] * c[0][r] + c[1][r] * c[1][r] +
                  c[2][r] * c[2][r] + c[3][r] * c[3][r];
        SWZ_ADD(s, SWZ_X1);  SWZ_ADD(q, SWZ_X1);
        SWZ_ADD(s, SWZ_X2);  SWZ_ADD(q, SWZ_X2);
        SWZ_ADD(s, SWZ_X4);  SWZ_ADD(q, SWZ_X4);
        SWZ_ADD(s, SWZ_X8);  SWZ_ADD(q, SWZ_X8);
        const float mu  = s * (1.0f / 64.0f);
        const float var = q * (1.0f / 64.0f) - mu * mu;
        mean[r] = mu;
        rstd[r] = __frsqrt_rn(var + 1e-5f);
    }

    // ---- normalize + ReLU, stage C-layout -> row-major LDS tile ----
    float* st = &lds[wave][0];
#pragma unroll
    for (int t = 0; t < 4; ++t) {
        const int col = t * 16 + m;
#pragma unroll
        for (int r = 0; r < 8; ++r) {
            float v = (c[t][r] - mean[r]) * rstd[r] * gv[t] + bev[t];
            v = fmaxf(v, 0.0f);
            st[(r + 8 * half) * LDS_STRIDE + col] = v;
        }
    }
    // same-wave LDS ops are in-order; compiler inserts s_wait_dscnt before use.

    // ---- GEMM2: bias[16x8] = h[16x64] * W2[64x8] (N padded to 16) ----
    v8f acc2 = {};
#pragma unroll
    for (int k = 0; k < 16; ++k) {
        const float* hp = &st[m * LDS_STRIDE + 4 * k + 2 * half];
        v2f av; av.x = hp[0]; av.y = hp[1];
        v2f bv; bv.x = w2a[k]; bv.y = w2b[k];
        acc2 = __builtin_amdgcn_wmma_f32_16x16x4_f32(false, av, false, bv,
                                                     (short)0, acc2, false, false);
    }

    // ---- store: D row r+8*half = j-in-tile, col m = head (only m<8 valid) ----
    if (m < 8) {
        const size_t base = (((size_t)b * 8 + m) * N_PTS + i) * (size_t)N_PTS
                            + (size_t)jt * 16 + 8 * half;
        float4 o0, o1;
        o0.x = acc2[0] + b2v; o0.y = acc2[1] + b2v;
        o0.z = acc2[2] + b2v; o0.w = acc2[3] + b2v;
        o1.x = acc2[4] + b2v; o1.y = acc2[5] + b2v;
        o1.z = acc2[6] + b2v; o1.w = acc2[7] + b2v;
        *(float4*)&out[base + 0] = o0;
        *(float4*)&out[base + 4] = o1;
    }
}

extern "C" void kernel_launch(void* const* d_in, const int* in_sizes, int n_in,
                              void* d_out, int out_size, void* d_ws, size_t ws_size,
                              hipStream_t stream) {
    const float* road = (const float*)d_in[0];
    const float* W1   = (const float*)d_in[1];
    const float* b1   = (const float*)d_in[2];
    const float* lng  = (const float*)d_in[3];
    const float* lnb  = (const float*)d_in[4];
    const float* W2   = (const float*)d_in[5];
    const float* b2   = (const float*)d_in[6];
    float* out = (float*)d_out;

    const int total_pts = 2 * N_PTS;                       // B*N
    const size_t need   = (size_t)total_pts * 4 * sizeof(float);

    dim3 grid(16384), block(256);   // 131072 wave-tiles, 8 waves/block
    if (ws_size >= need) {
        float4* table = (float4*)d_ws;
        hipLaunchKernelGGL(rpe_precompute, dim3((total_pts + 255) / 256),
                           dim3(256), 0, stream, road, table, total_pts);
        hipLaunchKernelGGL(spatial_rpe_kernel<true>, grid, block, 0, stream,
                           (const float*)table, W1, b1, lng, lnb, W2, b2, out);
    } else {
        hipLaunchKernelGGL(spatial_rpe_kernel<false>, grid, block, 0, stream,
                           road, W1, b1, lng, lnb, W2, b2, out);
    }
}